// SupervisedHMM_25477745999998
// MI455X (gfx1250) — compile-verified
//
#include <hip/hip_runtime.h>
#include <hip/hip_bf16.h>
#include <math.h>
#include <stdint.h>

// One wave32 per sequence (8 sequences per 256-thread block).
// Each wave streams its sequence's T int32 observations through LDS using
// CDNA5 async global->LDS B128 copies, pipelined 4 tiles deep (2 KB in
// flight per wave -> ~32 MB aggregate, above the 23.3 TB/s x ~1us BDP),
// synchronized ONLY with per-wave s_wait_asynccnt -- no workgroup barriers.
//
// Math (reference collapses to a scalar recurrence since alpha is broadcast):
//   out[b] = log( H * init_p * S^(T-1) * prod_t e_t ),  S = sum(trans_row)

#define WPB   8      // waves per block
#define TILE  128    // observations per wave tile: 32 lanes * 4 = 512 B
#define NBUF  4      // async pipeline depth (tiles in flight per wave)

__device__ __forceinline__ void async_ld16_to_lds(uint32_t lds_off, const int* gptr) {
  // GLOBAL_LOAD_ASYNC_TO_LDS_B128, GV form (64-bit per-lane vaddr).
  asm volatile("global_load_async_to_lds_b128 %0, %1, off"
               :: "v"(lds_off), "v"((unsigned long long)(uintptr_t)gptr)
               : "memory");
}

// Wait until at most min(rem,3) async ops remain outstanding; with in-order
// async completion this is exactly "tile t has landed".
__device__ __forceinline__ void wait_async_clamped(int rem) {
  if (rem >= 3)      asm volatile("s_wait_asynccnt 3" ::: "memory");
  else if (rem == 2) asm volatile("s_wait_asynccnt 2" ::: "memory");
  else if (rem == 1) asm volatile("s_wait_asynccnt 1" ::: "memory");
  else               asm volatile("s_wait_asynccnt 0" ::: "memory");
}

__global__ __launch_bounds__(256)
void hmm_fwd_kernel(const float* __restrict__ tp,   // (H,H)
                    const float* __restrict__ ep,   // (H,O)
                    const float* __restrict__ ip,   // (H,)
                    const int*   __restrict__ obs,  // (B,T)
                    const int*   __restrict__ lab,  // (B,)
                    float*       __restrict__ out,  // (B,)
                    int B, int T, int H, int O)
{
  __shared__ __align__(16) int obuf[WPB][NBUF][TILE]; // 16 KiB ring buffers
  __shared__ float etab[WPB][16];                     // per-wave emission row

  const int tid  = threadIdx.x;
  const int wave = tid >> 5;
  const int lane = tid & 31;
  // Pin the sequence index to an SGPR so parameter loads stay scalar.
  const int seqi = __builtin_amdgcn_readfirstlane((int)blockIdx.x * WPB + wave);
  if (seqi >= B) return;                              // no block barriers used

  const int l = lab[seqi];                            // s_load
  if (lane < O) etab[wave][lane] = ep[l * O + lane];

  float S = 0.0f;                                     // transition-row sum
  for (int h = 0; h < H; ++h) S += tp[l * H + h];
  const float initp = ip[l];
  __builtin_amdgcn_wave_barrier();                    // keep table fill ordered

  const int* seq    = obs + (size_t)seqi * (size_t)T;
  const int  ntiles = T / TILE;

  const uint32_t ldsw = (uint32_t)(uintptr_t)&obuf[wave][0][0];
  const uint32_t my16 = (uint32_t)lane * 16u;

  // Prologue: fill the pipeline (each lane copies exactly the 16 B it will
  // read, so a per-wave ASYNCcnt wait is the only ordering required).
  const int npro = (ntiles < NBUF) ? ntiles : NBUF;
  for (int k = 0; k < npro; ++k) {
    async_ld16_to_lds(ldsw + (uint32_t)k * (TILE * 4) + my16,
                      seq + (size_t)k * TILE + lane * 4);
  }

  float acc = 1.0f;                                   // running product of e_t
  for (int t = 0; t < ntiles; ++t) {
    wait_async_clamped(ntiles - 1 - t);               // tile t landed

    const int4* p = (const int4*)&obuf[wave][t & (NBUF - 1)][lane * 4];
    const int4 o = *p;                                // ds_load_b128
    const float e0 = etab[wave][o.x], e1 = etab[wave][o.y];
    const float e2 = etab[wave][o.z], e3 = etab[wave][o.w];
    acc *= (e0 * e1) * (e2 * e3);

    if (t + NBUF < ntiles) {                          // refill ring slot t&3
      async_ld16_to_lds(ldsw + (uint32_t)(t & (NBUF - 1)) * (TILE * 4) + my16,
                        seq + (size_t)(t + NBUF) * TILE + lane * 4);
    }
  }

  // Tail (T not a multiple of TILE): direct global gathers.
  for (int idx = ntiles * TILE + lane; idx < T; idx += 32) {
    acc *= etab[wave][seq[idx]];
  }

  // wave32 product reduction (intra-wave only; no LDS, no barrier).
  #pragma unroll
  for (int m = 16; m > 0; m >>= 1) acc *= __shfl_xor(acc, m, 32);

  if (lane == 0) {
    // S^(T-1) by square-and-multiply (keeps linear-space overflow semantics).
    float sp = 1.0f, bs = S;
    int n = T - 1;
    while (n > 0) { if (n & 1) sp *= bs; bs *= bs; n >>= 1; }
    out[seqi] = logf((float)H * initp * sp * acc);
  }
}

extern "C" void kernel_launch(void* const* d_in, const int* in_sizes, int n_in,
                              void* d_out, int out_size, void* d_ws, size_t ws_size,
                              hipStream_t stream) {
  const float* tp  = (const float*)d_in[0];  // transition_probs (H,H)
  const float* ep  = (const float*)d_in[1];  // emission_probs   (H,O)
  const float* ip  = (const float*)d_in[2];  // initial_probs    (H,)
  const int*   obs = (const int*)  d_in[3];  // observations     (B,T)
  const int*   lab = (const int*)  d_in[4];  // labels           (B,)
  // d_in[5] = sequence_length (device scalar); shapes derived from in_sizes.

  const int B = in_sizes[4];
  const int T = in_sizes[3] / B;
  const int H = in_sizes[2];
  const int O = in_sizes[1] / H;

  float* out = (float*)d_out;
  const int blocks = (B + WPB - 1) / WPB;
  hmm_fwd_kernel<<<blocks, 256, 0, stream>>>(tp, ep, ip, obs, lab, out, B, T, H, O);
}